// PrototypePatchMask_69913477644511
// MI455X (gfx1250) — compile-verified
//
#include <hip/hip_runtime.h>
#include <cstdint>

typedef float f4 __attribute__((ext_vector_type(4)));
typedef int   i4 __attribute__((ext_vector_type(4)));

#define NOISE_STD  0.2f
#define NOISE_MEAN 0.5f

#if defined(__AMDGCN__) && __has_builtin(__builtin_amdgcn_global_load_async_to_lds_b128)
#define HAVE_ASYNC_B128 1
typedef i4 __attribute__((address_space(1)))* as1_i4;
typedef i4 __attribute__((address_space(3)))* as3_i4;
#else
#define HAVE_ASYNC_B128 0
#endif

// ---------------------------------------------------------------------------
// Kernel 1: per-batch bounding rectangle of the nonzero 16x16 patch map,
// scaled to 512x512 pixel space (32x32 px per patch), widened by `widen`,
// clamped to [0,512). rect[b] = {h0, h1, w0, w1} (half-open).
// All-zero map -> full rectangle (matches argmax-of-all-False semantics).
// The 1 KB patch tile is staged into LDS with the gfx1250 async DMA path
// (global_load_async_to_lds_b128 + s_wait_asynccnt), then reduced.
// ---------------------------------------------------------------------------
__global__ __launch_bounds__(256) void patchmask_bbox(
    const int* __restrict__ patch, const int* __restrict__ widen,
    i4* __restrict__ rect) {
  __shared__ int s_minr, s_maxr, s_minc, s_maxc;
  const int b = blockIdx.x;
  const int t = threadIdx.x;                 // one thread per patch cell
  if (t == 0) { s_minr = 1 << 30; s_maxr = -1; s_minc = 1 << 30; s_maxc = -1; }

#if HAVE_ASYNC_B128
  __shared__ i4 tile4[64];                   // 256 ints = 16x16 patch map
  if (t < 64) {                              // waves 0-1 DMA the tile to LDS
    as1_i4 ga = (as1_i4)(uintptr_t)(patch + (b << 8) + (t << 2));
    as3_i4 la = (as3_i4)(uintptr_t)(&tile4[t]);
    __builtin_amdgcn_global_load_async_to_lds_b128(ga, la, 0, 0);
    __builtin_amdgcn_s_wait_asynccnt(0);     // issuing waves wait, then barrier
  }
  __syncthreads();                           // publishes tile + init values
  const int v = ((const int*)tile4)[t];
#else
  __syncthreads();
  const int v = patch[(b << 8) + t];
#endif

  if (v != 0) {
    atomicMin(&s_minr, t >> 4);
    atomicMax(&s_maxr, t >> 4);
    atomicMin(&s_minc, t & 15);
    atomicMax(&s_maxc, t & 15);
  }
  __syncthreads();
  if (t == 0) {
    const int wd = widen[0];
    i4 o;
    if (s_maxr < 0) {                        // no nonzero patch -> full mask
      o.x = 0; o.y = 512; o.z = 0; o.w = 512;
    } else {
      int h0 = s_minr * 32 - wd;       if (h0 < 0)   h0 = 0;
      int h1 = s_maxr * 32 + 32 + wd;  if (h1 > 512) h1 = 512;  // maxpix+wd+1
      int w0 = s_minc * 32 - wd;       if (w0 < 0)   w0 = 0;
      int w1 = s_maxc * 32 + 32 + wd;  if (w1 > 512) w1 = 512;
      o.x = h0; o.y = h1; o.z = w0; o.w = w1;
    }
    rect[b] = o;
  }
}

// ---------------------------------------------------------------------------
// Kernel 2: streaming composite. One thread per float4 (4 consecutive x).
// Load img only if the chunk touches the rectangle, noise only if it isn't
// fully inside -> ~402 MB instead of 604 MB of HBM traffic in the typical
// (rect == full image) case. Non-temporal: nothing is ever re-read, so do
// not displace the 192 MB L2 working set of anything else.
// ---------------------------------------------------------------------------
__global__ __launch_bounds__(256) void patchmask_apply(
    const f4* __restrict__ img, const f4* __restrict__ noise,
    const int* __restrict__ patch, const int* __restrict__ widen,
    const i4* __restrict__ rect, f4* __restrict__ out, int n4) {
  const int v = blockIdx.x * 256 + threadIdx.x;
  if (v >= n4) return;
  const int x  = (v & 127) << 2;   // 512 px / 4 per chunk = 128 chunks per row
  const int t  = v >> 7;           // y + 512*(c + 3*b)
  const int y  = t & 511;
  const int bc = t >> 9;           // c + 3*b
  const int b  = bc / 3;

  bool in0, in1, in2, in3;
  if (widen[0] > 0) {
    const i4 r = rect[b];          // {h0, h1, w0, w1}
    const bool yIn = (y >= r.x) & (y < r.y);
    in0 = yIn & (x     >= r.z) & (x     < r.w);
    in1 = yIn & (x + 1 >= r.z) & (x + 1 < r.w);
    in2 = yIn & (x + 2 >= r.z) & (x + 2 < r.w);
    in3 = yIn & (x + 3 >= r.z) & (x + 3 < r.w);
  } else {
    // widen == 0: raw upsampled patch mask (chunk never straddles a 32px block)
    const bool m = patch[(b << 8) + ((y >> 5) << 4) + (x >> 5)] != 0;
    in0 = in1 = in2 = in3 = m;
  }
  const bool anyIn = in0 | in1 | in2 | in3;
  const bool allIn = in0 & in1 & in2 & in3;

  f4 vi = {0.f, 0.f, 0.f, 0.f};
  f4 vn = {0.f, 0.f, 0.f, 0.f};
  if (anyIn)  vi = __builtin_nontemporal_load(&img[v]);
  if (!allIn) vn = __builtin_nontemporal_load(&noise[v]);

  f4 o;
  o.x = in0 ? vi.x : fminf(fmaxf(vn.x * NOISE_STD + NOISE_MEAN, 0.f), 1.f);
  o.y = in1 ? vi.y : fminf(fmaxf(vn.y * NOISE_STD + NOISE_MEAN, 0.f), 1.f);
  o.z = in2 ? vi.z : fminf(fmaxf(vn.z * NOISE_STD + NOISE_MEAN, 0.f), 1.f);
  o.w = in3 ? vi.w : fminf(fmaxf(vn.w * NOISE_STD + NOISE_MEAN, 0.f), 1.f);
  __builtin_nontemporal_store(o, &out[v]);
}

// ---------------------------------------------------------------------------
extern "C" void kernel_launch(void* const* d_in, const int* in_sizes, int n_in,
                              void* d_out, int out_size, void* d_ws, size_t ws_size,
                              hipStream_t stream) {
  const f4*  img   = (const f4*)d_in[0];   // [64,3,512,512] f32
  const int* patch = (const int*)d_in[1];  // [64,16,16] i32
  const f4*  noise = (const f4*)d_in[2];   // [64,3,512,512] f32
  const int* widen = (const int*)d_in[3];  // scalar i32 (=16)
  f4* out  = (f4*)d_out;
  i4* rect = (i4*)d_ws;                    // 64 * int4 = 1 KB scratch

  const int B = in_sizes[1] / 256;         // 64 batches
  patchmask_bbox<<<B, 256, 0, stream>>>(patch, widen, rect);

  const int n4 = out_size / 4;             // 12,582,912 float4 chunks
  const int blocks = (n4 + 255) / 256;
  patchmask_apply<<<blocks, 256, 0, stream>>>(img, noise, patch, widen, rect,
                                              out, n4);
}